// QLSTM_65481071403833
// MI455X (gfx1250) — compile-verified
//
#include <hip/hip_runtime.h>
#include <math.h>

typedef __attribute__((ext_vector_type(2))) float v2f;
typedef __attribute__((ext_vector_type(8))) float v8f;

#define T_LEN   1024
#define BATCH   256
#define HID     8
#define NQ      8
#define NL      2
#define NSTATES 256

// workspace layout (float offsets)
#define WS_GATES 0      // [0..3] = f, i, u(tanh gate), o
#define WS_CFIN  4      // final cx scalar
#define WS_V     16     // v[t] = o * tanh(c_t), 1024 floats
#define WS_M     2048   // M[t][h] = v_t * S_h + bp[h], 1024*8 floats

// ---------------------------------------------------------------------------
// Kernel 1: simulate the 4 param circuits (8 qubits, 256 real amplitudes).
// One block per gate; state double-buffered in LDS; <Z> on qubit 7 (bit 0).
// ---------------------------------------------------------------------------
__global__ void qlstm_gates_kernel(const float* __restrict__ pf,
                                   const float* __restrict__ pi,
                                   const float* __restrict__ pu,
                                   const float* __restrict__ po,
                                   float* __restrict__ ws) {
  __shared__ float buf[2][NSTATES];
  const int idx = threadIdx.x;      // state index, 0..255
  const int g   = blockIdx.x;       // 0=f,1=i,2=u,3=o
  const float* P = (g == 0) ? pf : (g == 1) ? pi : (g == 2) ? pu : po;

  int cur = 0;
  buf[0][idx] = (idx == 0) ? 1.0f : 0.0f;
  __syncthreads();

  for (int l = 0; l < NL; ++l) {
    // Ry layer: qubit q lives at bit position (7-q)
    for (int q = 0; q < NQ; ++q) {
      const float th = P[l * NQ + q] * 0.5f;
      const float c = cosf(th), s = sinf(th);
      const int mask = 1 << (7 - q);
      const float a0 = buf[cur][idx & ~mask];
      const float a1 = buf[cur][idx |  mask];
      const float nv = (idx & mask) ? (s * a0 + c * a1) : (c * a0 - s * a1);
      buf[cur ^ 1][idx] = nv;
      __syncthreads();
      cur ^= 1;
    }
    // CNOT ladder (0->1, ..., 6->7) then wrap (7->0)
    for (int q = 0; q < NQ; ++q) {
      const int ctrl = (q < 7) ? q : 7;
      const int tgt  = (q < 7) ? q + 1 : 0;
      const int cmask = 1 << (7 - ctrl);
      const int tmask = 1 << (7 - tgt);
      const int src = (idx & cmask) ? (idx ^ tmask) : idx;
      const float nv = buf[cur][src];
      buf[cur ^ 1][idx] = nv;
      __syncthreads();
      cur ^= 1;
    }
  }

  // <Z> on last qubit (bit 0): +prob(bit0==0) - prob(bit0==1)
  const float a = buf[cur][idx];
  const float val = ((idx & 1) ? -1.0f : 1.0f) * a * a;
  __syncthreads();
  buf[0][idx] = val;
  __syncthreads();
  for (int off = 128; off > 0; off >>= 1) {
    if (idx < off) buf[0][idx] += buf[0][idx + off];
    __syncthreads();
  }
  if (idx == 0) {
    const float E = buf[0][0];
    const float r = (g == 2) ? tanhf(E) : 1.0f / (1.0f + expf(-E));
    ws[WS_GATES + g] = r;
  }
}

// ---------------------------------------------------------------------------
// Kernel 2: serial scalar recurrence c_t = f*c + i*g ; v[t] = o * tanh(c_t)
// ---------------------------------------------------------------------------
__global__ void qlstm_seq_kernel(float* __restrict__ ws) {
  if (threadIdx.x != 0 || blockIdx.x != 0) return;
  const float f = ws[WS_GATES + 0];
  const float i = ws[WS_GATES + 1];
  const float u = ws[WS_GATES + 2];
  const float o = ws[WS_GATES + 3];
  const float ig = i * u;
  float c = 0.0f;
  for (int t = 0; t < T_LEN; ++t) {
    c = f * c + ig;
    ws[WS_V + t] = o * tanhf(c);
  }
  ws[WS_CFIN] = c;
}

// ---------------------------------------------------------------------------
// Kernel 3: M[t,h] = v_t * S_h + bp[h] via V_WMMA_F32_16X16X4_F32.
// One wave per 16-row tile of t (64 tiles). Rank-1: only K=0 populated
// (A vgpr0 lanes 0-15 = v_t, B vgpr0 lanes 0-7 = S_h), so the result is
// layout-robust. bp added post-WMMA using the documented C/D layout.
// ---------------------------------------------------------------------------
__global__ void qlstm_wmma_kernel(const float* __restrict__ Wp,
                                  const float* __restrict__ bp,
                                  float* __restrict__ ws) {
  const int lane = threadIdx.x & 31;
  const int wave = (int)(blockIdx.x * blockDim.x + threadIdx.x) >> 5;
  const int t0 = wave * 16;
  const float* v = ws + WS_V;
  float* M = ws + WS_M;

  v2f a; a.x = 0.0f; a.y = 0.0f;            // A: 16x4, column K=0 = v_t
  if (lane < 16) a.x = v[t0 + lane];

  v2f b; b.x = 0.0f; b.y = 0.0f;            // B: 4x16, row K=0 = S_n (n<8)
  if (lane < 8) {
    float s = 0.0f;
#pragma unroll
    for (int q = 0; q < 8; ++q) s += Wp[lane * 8 + q];
    b.x = s;
  }

  v8f c = {0.0f, 0.0f, 0.0f, 0.0f, 0.0f, 0.0f, 0.0f, 0.0f};
  // (neg_a, A, neg_b, B, c_mod, C, reuse_a, reuse_b)
  v8f d = __builtin_amdgcn_wmma_f32_16x16x4_f32(false, a, false, b,
                                                (short)0, c, false, false);

  // C/D layout: VGPR r -> lanes 0-15: (M=r, N=lane); lanes 16-31: (M=r+8, N=lane-16)
  int n = -1, mbase = 0;
  if (lane < 8)                      { n = lane;      mbase = t0;     }
  else if (lane >= 16 && lane < 24)  { n = lane - 16; mbase = t0 + 8; }
  if (n >= 0) {
    const float bpv = bp[n];
#pragma unroll
    for (int r = 0; r < 8; ++r) {
      M[(mbase + r) * 8 + n] = d[r] + bpv;
    }
  }
}

// ---------------------------------------------------------------------------
// Kernel 4: bandwidth-bound broadcast fill.
// blocks 0..1023: outs[t,b,:] = M[t,:]   (coalesced 32B/lane float4 stores)
// block 1024:     hx[b,:]  = M[1023,:]
// block 1025:     cx[b,:]  = c_final
// ---------------------------------------------------------------------------
__global__ void qlstm_fill_kernel(float* __restrict__ out,
                                  const float* __restrict__ ws) {
  const int b = threadIdx.x;
  const int blk = blockIdx.x;
  const float* M = ws + WS_M;

  if (blk < T_LEN) {
    const float4* mrow = (const float4*)(M + blk * HID);
    const float4 m0 = mrow[0], m1 = mrow[1];
    float4* dst = (float4*)(out + (size_t)blk * (BATCH * HID) + b * HID);
    dst[0] = m0; dst[1] = m1;
  } else if (blk == T_LEN) {
    const float4* mrow = (const float4*)(M + (T_LEN - 1) * HID);
    const float4 m0 = mrow[0], m1 = mrow[1];
    float4* dst = (float4*)(out + (size_t)T_LEN * BATCH * HID + b * HID);
    dst[0] = m0; dst[1] = m1;
  } else {
    const float cf = ws[WS_CFIN];
    const float4 cv = {cf, cf, cf, cf};
    float4* dst = (float4*)(out + (size_t)T_LEN * BATCH * HID + BATCH * HID + b * HID);
    dst[0] = cv; dst[1] = cv;
  }
}

// ---------------------------------------------------------------------------
extern "C" void kernel_launch(void* const* d_in, const int* in_sizes, int n_in,
                              void* d_out, int out_size, void* d_ws, size_t ws_size,
                              hipStream_t stream) {
  // setup_inputs order:
  // 0 inputs (DEAD - never read), 1 Wf, 2 bf, 3 Wi, 4 bi, 5 Wu, 6 bu,
  // 7 Wo, 8 bo, 9 pf, 10 pi, 11 pu, 12 po, 13 Wp, 14 bp
  const float* pf = (const float*)d_in[9];
  const float* pi = (const float*)d_in[10];
  const float* pu = (const float*)d_in[11];
  const float* po = (const float*)d_in[12];
  const float* Wp = (const float*)d_in[13];
  const float* bp = (const float*)d_in[14];
  float* ws  = (float*)d_ws;
  float* out = (float*)d_out;

  qlstm_gates_kernel<<<4, NSTATES, 0, stream>>>(pf, pi, pu, po, ws);
  qlstm_seq_kernel<<<1, 32, 0, stream>>>(ws);
  qlstm_wmma_kernel<<<2, 1024, 0, stream>>>(Wp, bp, ws);   // 64 waves, 1 WMMA each
  qlstm_fill_kernel<<<T_LEN + 2, BATCH, 0, stream>>>(out, ws);
}